// BernsteinSplineCouplingBlock_54339926229632
// MI455X (gfx1250) — compile-verified
//
#include <hip/hip_runtime.h>
#include <math.h>

// ---- problem constants (from reference) ----
#define CH      64          // CHANNELS
#define SPLITN  32          // SPLIT1 == SPLIT2
#define DEG     10          // DEGREE
#define ODIM    384         // SPLIT2 * (DEGREE + 2)
#define BROWS   32          // batch rows per workgroup
#define LDSTR   388         // LDS row stride (floats), 388 % 64 == 4 -> half-wave groups on disjoint banks

typedef float v2f __attribute__((ext_vector_type(2)));
typedef float v8f __attribute__((ext_vector_type(8)));

__device__ __forceinline__ float softplus_f(float v) {
    // stable log(1 + exp(v))
    if (v > 20.0f)  return v;
    if (v < -20.0f) return __expf(v);
    return log1pf(__expf(v));
}

__device__ __forceinline__ float signf(float v) {
    return (v > 0.0f) ? 1.0f : ((v < 0.0f) ? -1.0f : 0.0f);
}

__global__ __launch_bounds__(256)
void bernstein_coupling_kernel(const float* __restrict__ x,
                               const float* __restrict__ W,
                               const float* __restrict__ b,
                               float* __restrict__ out)
{
    __shared__ float s_s[BROWS * LDSTR];   // s = x1 @ W^T + b, [32 x 384] (+pad)

    const int tid  = threadIdx.x;
    const int lane = tid & 31;
    const int wave = tid >> 5;
    const int half = lane >> 4;     // 0: lanes 0-15, 1: lanes 16-31
    const int mn   = lane & 15;     // M (A) / N (B,C,D) index within tile
    const int rowBase = blockIdx.x * BROWS;

    // ---------------- Phase 1: GEMM via V_WMMA_F32_16X16X4_F32 ----------------
    // s[n,o] = sum_{k<32} x[n,k] * W[o,k] + b[o]
    #pragma unroll
    for (int bt = 0; bt < BROWS / 16; ++bt) {
        // A fragment: 16x32 slab of x1, as 8 K-steps of 16x4.
        // Lane layout per step s: lanes 0-15 hold K = 4s,4s+1 ; lanes 16-31 hold K = 4s+2,4s+3 (M = mn).
        const int arow = rowBase + bt * 16 + mn;
        v2f areg[8];
        #pragma unroll
        for (int s = 0; s < 8; ++s)
            areg[s] = *(const v2f*)&x[arow * CH + 4 * s + 2 * half];

        for (int nt = wave; nt < ODIM / 16; nt += 8) {
            const int o = nt * 16 + mn;          // output column for this lane
            const float bv = b[o];
            v8f acc;
            #pragma unroll
            for (int g = 0; g < 8; ++g) acc[g] = bv;   // bias folded into C

            #pragma unroll
            for (int s = 0; s < 8; ++s) {
                // B fragment 4x16: B[k][n] = W[o=n][k]; same half-wave K split as A.
                v2f breg = *(const v2f*)&W[o * SPLITN + 4 * s + 2 * half];
                acc = __builtin_amdgcn_wmma_f32_16x16x4_f32(
                        /*neg_a=*/false, areg[s],
                        /*neg_b=*/false, breg,
                        /*c_mod=*/(short)0, acc,
                        /*reuse_a=*/false, /*reuse_b=*/false);
            }

            // D layout: VGPR g -> (M = g + 8*half, N = mn)
            #pragma unroll
            for (int g = 0; g < 8; ++g)
                s_s[(bt * 16 + g + 8 * half) * LDSTR + o] = acc[g];
        }
    }
    __syncthreads();

    // ---------------- Phase 2: spline evaluation (1024 elems / block) ----------------
    for (int e = tid; e < BROWS * SPLITN; e += 256) {
        const int nl = e >> 5;                    // local batch row
        const int j  = e & 31;                    // spline channel
        const float* sp = &s_s[nl * LDSTR + j * (DEG + 2)];

        // cumsum of softplus -> monotone Bernstein coeffs, normalized
        float coeffs[DEG + 1];
        coeffs[0] = 0.0f;
        float run = 0.0f;
        #pragma unroll
        for (int c = 0; c < DEG; ++c) {
            run += softplus_f(sp[c]);
            coeffs[c + 1] = run;
        }
        const float inv = 1.0f / coeffs[DEG];
        #pragma unroll
        for (int c = 0; c <= DEG; ++c) coeffs[c] *= inv;

        float wv = softplus_f(sp[DEG]);
        wv += 0.1f * signf(wv);
        float hv = sp[DEG + 1];
        hv += 0.1f * signf(hv);

        const float x2 = x[(rowBase + nl) * CH + SPLITN + j];
        const float t  = x2 / wv + 0.5f;
        const float tc = fminf(fmaxf(t, 0.0f), 1.0f);
        const float omt = 1.0f - tc;

        // derivative: DEG * DeCasteljau(tc, diff(coeffs))  (10 coeffs, 9 steps)
        float dv[DEG];
        #pragma unroll
        for (int c = 0; c < DEG; ++c) dv[c] = coeffs[c + 1] - coeffs[c];
        #pragma unroll
        for (int it = DEG - 1; it > 0; --it)
            for (int c = 0; c < it; ++c)
                dv[c] = omt * dv[c] + tc * dv[c + 1];
        const float deriv = (float)DEG * dv[0];

        // value: DeCasteljau(tc, coeffs)  (11 coeffs, 10 steps)
        #pragma unroll
        for (int it = DEG; it > 0; --it)
            for (int c = 0; c < it; ++c)
                coeffs[c] = omt * coeffs[c] + tc * coeffs[c + 1];
        const float ym = coeffs[0];

        float y = (t < 0.0f) ? (t * deriv)
                : (t > 1.0f) ? (1.0f + (t - 1.0f) * deriv)
                             : ym;
        y = (y - 0.5f) * hv;

        out[(rowBase + nl) * CH + SPLITN + j] = y;
    }

    // ---------------- Phase 3: pass-through copy of x1 (float4) ----------------
    {
        const int r4 = tid >> 3;   // 0..31 local row
        const int c4 = tid & 7;    // 0..7  float4 column
        const int n  = rowBase + r4;
        *(float4*)&out[n * CH + c4 * 4] = *(const float4*)&x[n * CH + c4 * 4];
    }
}

extern "C" void kernel_launch(void* const* d_in, const int* in_sizes, int n_in,
                              void* d_out, int out_size, void* d_ws, size_t ws_size,
                              hipStream_t stream) {
    (void)n_in; (void)d_ws; (void)ws_size; (void)out_size;
    const float* x = (const float*)d_in[0];   // [BATCH, 64]
    const float* W = (const float*)d_in[1];   // [384, 32]
    const float* b = (const float*)d_in[2];   // [384]
    float* out = (float*)d_out;               // [BATCH, 64]

    const int batch   = in_sizes[0] / CH;     // 65536
    const int nblocks = batch / BROWS;        // 2048

    bernstein_coupling_kernel<<<nblocks, 256, 0, stream>>>(x, W, b, out);
}